// LightGCNLayer_63136019251354
// MI455X (gfx1250) — compile-verified
//
#include <hip/hip_runtime.h>
#include <stdint.h>

// out = A_coo @ x : out[r] += v * x[c]
// N=100000 nodes, E=3200000 edges, D=64.
// CSR build (histogram -> scan -> bin) in workspace, then one wave32 per row
// accumulates in registers -> single coalesced 256B store per row.
// Row payload chunks are staged into LDS via gfx1250 async loads
// (global_load_async_to_lds_b64 + s_wait_asynccnt), software-pipelined
// across the wave's strided row list.

static constexpr int EMBED = 64;

// ---------------- phase 0: zero counts ----------------
__global__ void k_zero_u32(uint32_t* __restrict__ p, int n) {
  int i = blockIdx.x * blockDim.x + threadIdx.x;
  if (i < n) p[i] = 0u;
}

// ---------------- phase 1: histogram of rows ----------------
__global__ void k_hist(const int* __restrict__ rows, uint32_t* __restrict__ counts, int E) {
  int e = blockIdx.x * blockDim.x + threadIdx.x;
  if (e < E) atomicAdd(&counts[rows[e]], 1u);
}

// ---------------- phase 2: single-block exclusive scan (in place) ----------------
__global__ __launch_bounds__(1024)
void k_scan(uint32_t* __restrict__ offs,      // in: counts[N], out: offsets[N+1]
            uint32_t* __restrict__ cursor,    // out: copy of offsets[0..N-1]
            int n, uint32_t total) {
  __shared__ uint32_t sdata[1024];
  __shared__ uint32_t sbase;
  if (threadIdx.x == 0) sbase = 0u;
  __syncthreads();
  for (int chunk = 0; chunk < n; chunk += 1024) {
    int i = chunk + (int)threadIdx.x;
    uint32_t v = (i < n) ? offs[i] : 0u;
    sdata[threadIdx.x] = v;
    __syncthreads();
    for (int d = 1; d < 1024; d <<= 1) {
      uint32_t t = (threadIdx.x >= (unsigned)d) ? sdata[threadIdx.x - d] : 0u;
      __syncthreads();
      sdata[threadIdx.x] += t;
      __syncthreads();
    }
    uint32_t incl = sdata[threadIdx.x];
    uint32_t excl = incl - v;
    uint32_t base = sbase;
    if (i < n) {
      uint32_t o = base + excl;
      offs[i] = o;
      cursor[i] = o;
    }
    __syncthreads();
    if (threadIdx.x == 1023) sbase = base + incl;
    __syncthreads();
  }
  if (threadIdx.x == 0) offs[n] = total;
}

// ---------------- phase 3: bin (col,val) by row ----------------
__global__ void k_scatter(const int* __restrict__ rows, const int* __restrict__ cols,
                          const float* __restrict__ vals, uint32_t* __restrict__ cursor,
                          uint2* __restrict__ payload, int E) {
  int e = blockIdx.x * blockDim.x + threadIdx.x;
  if (e < E) {
    int r = rows[e];
    uint32_t slot = atomicAdd(&cursor[r], 1u);
    uint2 p;
    p.x = (uint32_t)cols[e];
    p.y = __float_as_uint(vals[e]);
    payload[slot] = p;               // global_store_b64
  }
}

// ---------------- phase 4: one wave32 per row, async-pipelined ----------------
// Each wave owns rows {w, w+stride, ...}. While computing row k, the async
// unit copies row k+stride's first 32-edge chunk (256B) into LDS.
__global__ __launch_bounds__(256)
void k_gather(const float2* __restrict__ x2, const uint32_t* __restrict__ offs,
              const uint2* __restrict__ payload, float2* __restrict__ out2,
              int nrows, int E) {
  __shared__ uint2 abuf[8][2][32];   // 8 waves x 2 stages x 32 edges = 4KB
  const int lane = (int)(threadIdx.x & 31u);
  const int wsl  = (int)(threadIdx.x >> 5);
  const int stride = (int)gridDim.x * 8;
  int row = (int)blockIdx.x * 8 + wsl;
  if (row >= nrows) return;          // wave-uniform: EXEC stays all-ones below

  const uint32_t Eu = (uint32_t)E;

  // Issue async copy of up to 32 edges starting at beg into stage s.
  // Indices are clamped (not predicated) so EXEC remains all-ones.
  auto issue = [&](uint32_t beg, uint32_t end, int s) {
    uint32_t idx = beg + (uint32_t)lane;
    uint32_t hi  = (end > beg) ? (end - 1u) : beg;
    if (idx > hi) idx = hi;
    if (idx >= Eu) idx = Eu - 1u;
    const uint2* gp = payload + idx;                            // 64-bit VGPR pair
    uint32_t ldsa = (uint32_t)(uintptr_t)(&abuf[wsl][s][lane]); // LDS byte addr
    asm volatile("global_load_async_to_lds_b64 %0, %1, off"
                 :: "v"(ldsa), "v"(gp)
                 : "memory");
  };

  // Fully-unrolled common case (cnt == 32): lets the compiler clause the 32
  // coalesced b64 x-gathers and hide L2 latency behind the FMA chain.
  auto accum_full = [&](uint2 p, float2& acc) {
#pragma unroll
    for (int j = 0; j < 32; ++j) {
      uint32_t c = __shfl(p.x, j, 32);
      float    v = __uint_as_float(__shfl(p.y, j, 32));
      float2 xv = x2[(size_t)c * 32 + lane];
      acc.x = fmaf(v, xv.x, acc.x);
      acc.y = fmaf(v, xv.y, acc.y);
    }
  };
  auto accum_part = [&](uint2 p, int cnt, float2& acc) {
    for (int j = 0; j < cnt; ++j) {
      uint32_t c = __shfl(p.x, j, 32);
      float    v = __uint_as_float(__shfl(p.y, j, 32));
      float2 xv = x2[(size_t)c * 32 + lane];
      acc.x = fmaf(v, xv.x, acc.x);
      acc.y = fmaf(v, xv.y, acc.y);
    }
  };

  int stage = 0;
  uint32_t begA = offs[row];
  uint32_t endA = offs[row + 1];
  issue(begA, endA, stage);          // prime the pipeline (ASYNCcnt = 1)

  while (row < nrows) {
    const int nrow = row + stride;
    const bool haveNext = (nrow < nrows);
    uint32_t begB = 0u, endB = 0u;
    if (haveNext) {
      begB = offs[nrow];
      endB = offs[nrow + 1];
      issue(begB, endB, stage ^ 1);                        // ASYNCcnt = 2
      asm volatile("s_wait_asynccnt 0x1" ::: "memory");    // current chunk done
    } else {
      asm volatile("s_wait_asynccnt 0x0" ::: "memory");    // drain
    }

    float2 acc; acc.x = 0.0f; acc.y = 0.0f;

    // Chunk 0 of this row comes from LDS (async-copied): ds_load_b64.
    {
      uint2 p = abuf[wsl][stage][lane];
      const uint32_t deg0 = endA - begA;
      if (deg0 >= 32u) accum_full(p, acc);
      else             accum_part(p, (int)deg0, acc);
    }
    // Rare tail chunks (degree > 32): direct coalesced b64 loads.
    for (uint32_t base = begA + 32u; base < endA; base += 32u) {
      uint32_t idx = base + (uint32_t)lane;
      uint2 p; p.x = 0u; p.y = 0u;
      if (idx < endA) p = payload[idx];
      if (base + 32u < endA)
        __builtin_prefetch(&payload[base + 32u + (uint32_t)lane], 0, 0);
      const uint32_t rem = endA - base;
      if (rem >= 32u) accum_full(p, acc);
      else            accum_part(p, (int)rem, acc);
    }

    out2[(size_t)row * 32 + lane] = acc;   // single coalesced 256B store

    row = nrow;
    begA = begB; endA = endB;
    stage ^= 1;
  }
}

// ---------------- fallback: direct atomic scatter (if ws too small) ----------------
__global__ void k_zero_out(float4* __restrict__ o, int n4) {
  int i = blockIdx.x * blockDim.x + threadIdx.x;
  if (i < n4) { float4 z; z.x = z.y = z.z = z.w = 0.0f; o[i] = z; }
}

__global__ __launch_bounds__(256)
void k_atomic_scatter(const float2* __restrict__ x2, const int* __restrict__ rows,
                      const int* __restrict__ cols, const float* __restrict__ vals,
                      float* __restrict__ out, int E) {
  const int lane = (int)(threadIdx.x & 31u);
  const int e = (blockIdx.x * ((int)blockDim.x >> 5)) + ((int)threadIdx.x >> 5);
  if (e >= E) return;
  const int r = rows[e];
  const int c = cols[e];
  const float v = vals[e];
  float2 xv = x2[(size_t)c * 32 + lane];
  atomicAdd(&out[(size_t)r * EMBED + 2 * lane],     v * xv.x);
  atomicAdd(&out[(size_t)r * EMBED + 2 * lane + 1], v * xv.y);
}

extern "C" void kernel_launch(void* const* d_in, const int* in_sizes, int n_in,
                              void* d_out, int out_size, void* d_ws, size_t ws_size,
                              hipStream_t stream) {
  const float* x    = (const float*)d_in[0];
  const int*   rows = (const int*)d_in[1];
  const int*   cols = (const int*)d_in[2];
  const float* vals = (const float*)d_in[3];
  float* out = (float*)d_out;

  const int E = in_sizes[1];
  const int N = out_size / EMBED;

  const size_t payloadBytes = (size_t)E * sizeof(uint2);
  const size_t offsBytes    = (size_t)(N + 1) * sizeof(uint32_t);
  const size_t curBytes     = (size_t)N * sizeof(uint32_t);
  const size_t need         = payloadBytes + offsBytes + curBytes;

  if (ws_size >= need) {
    uint2*    payload = (uint2*)d_ws;
    uint32_t* offs    = (uint32_t*)((char*)d_ws + payloadBytes);
    uint32_t* cursor  = (uint32_t*)((char*)d_ws + payloadBytes + offsBytes);

    k_zero_u32<<<dim3((N + 1 + 255) / 256), dim3(256), 0, stream>>>(offs, N + 1);
    k_hist<<<dim3((E + 255) / 256), dim3(256), 0, stream>>>(rows, offs, E);
    k_scan<<<dim3(1), dim3(1024), 0, stream>>>(offs, cursor, N, (uint32_t)E);
    k_scatter<<<dim3((E + 255) / 256), dim3(256), 0, stream>>>(rows, cols, vals, cursor, payload, E);

    // 3125 blocks x 8 waves = 25000 waves; each wave pipelines 4 rows.
    const int blocks = 3125;
    k_gather<<<dim3(blocks), dim3(256), 0, stream>>>(
        (const float2*)x, offs, payload, (float2*)out, N, E);
  } else {
    const int n4 = out_size / 4;
    k_zero_out<<<dim3((n4 + 255) / 256), dim3(256), 0, stream>>>((float4*)out, n4);
    k_atomic_scatter<<<dim3((E + 7) / 8), dim3(256), 0, stream>>>(
        (const float2*)x, rows, cols, vals, out, E);
  }
}